// CrossAttention_39109972197429
// MI455X (gfx1250) — compile-verified
//
#include <hip/hip_runtime.h>
#include <math.h>

// MI455X / gfx1250, wave32. All matmuls via V_WMMA_F32_16X16X4_F32 (exact f32).
typedef __attribute__((ext_vector_type(2))) float v2f;
typedef __attribute__((ext_vector_type(8))) float v8f;

#define N_SP   16384   // H*W
#define WIDTH  128
#define CCH    256
#define OCH    512

__device__ __forceinline__ v8f wmma4(v2f a, v2f b, v8f c) {
  // 8 args: (neg_a, A, neg_b, B, c_mod, C, reuse_a, reuse_b)
  return __builtin_amdgcn_wmma_f32_16x16x4_f32(false, a, false, b, (short)0, c,
                                               false, false);
}

// ---------------------------------------------------------------------------
// Kernel 1: QKV = W(512x256) * X(256x16384) + bias, tiled 64(M) x 128(N).
// Double-buffered LDS + register pipelining: one barrier per K-chunk, global
// loads for chunk kk+1 issued before computing chunk kk, distance-2
// global_prefetch_b8 on the streamed X rows.
// M-tile 64 == one head's {q[0:16], k[16:32], v[32:64]} rows, so the block
// also computes its n-chunk's partial E = K * V^T (16x32) and writes it to a
// deterministic per-block partial slot (no float atomics -> bitwise stable).
// grid = (128 n-tiles, 8 heads, 8 = tensor*4 + batch), block = 256 (8 waves)
// ---------------------------------------------------------------------------
__global__ __launch_bounds__(256) void qkv_kvred_kernel(
    const float* __restrict__ rgb, const float* __restrict__ chm,
    const float* __restrict__ rw,  const float* __restrict__ rb,
    const float* __restrict__ cw,  const float* __restrict__ cb,
    float* __restrict__ qkv, float* __restrict__ part)
{
  __shared__ float lA[2][64 * 20];    // W tile  [m][k], stride 20: conflict-free
  __shared__ float lB[2][16 * 136];   // X tile  [k][n], stride 136: halves disjoint
  __shared__ float lKV[48 * 136];     // K/V rows of the output tile [m-16][n]

  const int z     = blockIdx.z;          // tensor*4 + batch
  const int input = z >> 2;
  const int b     = z & 3;
  const int h     = blockIdx.y;
  const int n0    = blockIdx.x * 128;
  const int m0    = h * 64;

  const float* X    = (input ? chm : rgb) + (size_t)b * CCH * N_SP;
  const float* W    = input ? cw : rw;
  const float* Bias = input ? cb : rb;

  const int tid  = threadIdx.x;
  const int lane = tid & 31;
  const int lh   = lane >> 4;            // lane half (K = 2*lh + j for A/B frags)
  const int l16  = lane & 15;
  const int wid  = tid >> 5;
  const int mw   = wid & 3;              // wave row group: 16*mw
  const int nw   = wid >> 2;             // wave col group: 64*nw

  v8f acc0 = {}, acc1 = {}, acc2 = {}, acc3 = {};

  const int arow = tid >> 2;             // 0..63
  const int acol = (tid & 3) * 4;        // 0,4,8,12
  const int brow = tid >> 5;             // 0..7
  const int bcol = (tid & 31) * 4;       // 0..124

  const float* wp  = &W[(size_t)(m0 + arow) * CCH + acol];
  const float* xp0 = &X[(size_t)brow * N_SP + n0 + bcol];

  // pipeline preload: K-chunk 0
  float4 avr  = *(const float4*)&wp[0];
  float4 bvr0 = *(const float4*)&xp0[0];
  float4 bvr1 = *(const float4*)&xp0[(size_t)8 * N_SP];

  for (int kk = 0; kk < 16; ++kk) {
    const int buf = kk & 1;
    *(float4*)&lA[buf][arow * 20 + acol]        = avr;
    *(float4*)&lB[buf][brow * 136 + bcol]       = bvr0;
    *(float4*)&lB[buf][(brow + 8) * 136 + bcol] = bvr1;
    __syncthreads();                       // single barrier per chunk (see header)

    if (kk < 15) {                         // issue next chunk's loads now,
      const int k0 = (kk + 1) * 16;        // consumed at top of next iteration
      avr  = *(const float4*)&wp[k0];
      bvr0 = *(const float4*)&xp0[(size_t)k0 * N_SP];
      bvr1 = *(const float4*)&xp0[(size_t)(k0 + 8) * N_SP];
    }
    if (kk < 14) {                         // distance-2 prefetch of streamed X
      const size_t k0p = (size_t)(kk + 2) * 16;
      __builtin_prefetch(&xp0[k0p * N_SP], 0, 0);
      __builtin_prefetch(&xp0[(k0p + 8) * N_SP], 0, 0);
    }

    const float* la = lA[buf];
    const float* lb = lB[buf];
#pragma unroll
    for (int ks = 0; ks < 4; ++ks) {
      const int kb = ks * 4 + 2 * lh;
      const v2f a = *(const v2f*)&la[(16 * mw + l16) * 20 + kb];  // ds_load_b64
      const float* bp = &lb[kb * 136 + 64 * nw + l16];
      v2f bf;
      bf.x = bp[0];  bf.y = bp[136];       acc0 = wmma4(a, bf, acc0);
      bf.x = bp[16]; bf.y = bp[136 + 16];  acc1 = wmma4(a, bf, acc1);
      bf.x = bp[32]; bf.y = bp[136 + 32];  acc2 = wmma4(a, bf, acc2);
      bf.x = bp[48]; bf.y = bp[136 + 48];  acc3 = wmma4(a, bf, acc3);
    }
  }
  __syncthreads();

  // bias, store QKV, and stage K/V rows (local m 16..63) into LDS
  float* outp = qkv + (size_t)z * OCH * N_SP;
#pragma unroll
  for (int r = 0; r < 8; ++r) {
    const int mloc = 16 * mw + r + 8 * lh;            // 0..63 within tile
    const int mrow = m0 + mloc;
    const float bias = Bias[mrow];
    const float vs[4] = {acc0[r], acc1[r], acc2[r], acc3[r]};
#pragma unroll
    for (int s = 0; s < 4; ++s) {
      const int ncol = 64 * nw + 16 * s + l16;
      const float v = vs[s] + bias;
      outp[(size_t)mrow * N_SP + n0 + ncol] = v;
      if (mw >= 1) lKV[(mloc - 16) * 136 + ncol] = v;
    }
  }
  __syncthreads();

  // partial E[k][d] = sum_{n in tile} K[k,n] * V[d,n]  (two waves, 64 WMMA)
  if (wid < 2) {
    v8f e = {};
    const int dbase = wid * 16;
#pragma unroll 4
    for (int ks = 0; ks < 32; ++ks) {
      const int nb = ks * 4 + 2 * lh;
      const v2f a  = *(const v2f*)&lKV[l16 * 136 + nb];                 // K rows
      const v2f bv = *(const v2f*)&lKV[(16 + dbase + l16) * 136 + nb];  // V^T cols
      e = wmma4(a, bv, e);
    }
    float* pp = part + (((size_t)z * 8 + h) * 128 + blockIdx.x) * 512;
#pragma unroll
    for (int r = 0; r < 8; ++r)
      pp[(r + 8 * lh) * 32 + dbase + l16] = e[r];
  }
}

// ---------------------------------------------------------------------------
// Kernel 2: reduce 128 partials per (tensor,b,head) and softmax over d (32).
// grid = 64, block = 256
// ---------------------------------------------------------------------------
__global__ __launch_bounds__(256) void reduce_softmax_kernel(
    const float* __restrict__ part, float* __restrict__ attn)
{
  __shared__ float e[16 * 33];
  const int g   = blockIdx.x;            // z*8 + h
  const int tid = threadIdx.x;
  const float* pb = part + (size_t)g * 128 * 512;
  float s0 = 0.f, s1 = 0.f;
  for (int nb = 0; nb < 128; ++nb) {
    s0 += pb[(size_t)nb * 512 + tid];
    s1 += pb[(size_t)nb * 512 + 256 + tid];
  }
  e[(tid >> 5) * 33 + (tid & 31)]       = s0;   // k = tid/32
  e[((tid >> 5) + 8) * 33 + (tid & 31)] = s1;   // k = 8 + tid/32
  __syncthreads();
  if (tid < 16) {
    float m = -1e30f;
    for (int d = 0; d < 32; ++d) m = fmaxf(m, e[tid * 33 + d]);
    float ex[32]; float sum = 0.f;
    for (int d = 0; d < 32; ++d) { ex[d] = expf(e[tid * 33 + d] - m); sum += ex[d]; }
    const float inv = 1.0f / sum;
    float* ap = attn + (size_t)g * 512 + tid * 32;
    for (int d = 0; d < 32; ++d) ap[d] = ex[d] * inv;
  }
}

// ---------------------------------------------------------------------------
// Kernel 3: out[b, 32h+d, y, x] = (c_attn^T rq)[d,n] + (r_attn^T cq)[d,n]
//                                 + dw3x3(rv) + dw3x3(cv) + biases
// grid = (128 rows, 8 heads, 4 batches), block = 256 (8 waves x 16 cols = W)
// All V data (134 MB) fits in the 192 MB L2, so the 3x3 halo re-reads hit L2.
// ---------------------------------------------------------------------------
__global__ __launch_bounds__(256) void output_kernel(
    const float* __restrict__ qkv, const float* __restrict__ attn,
    const float* __restrict__ rpw, const float* __restrict__ rpb,
    const float* __restrict__ cpw, const float* __restrict__ cpb,
    float* __restrict__ out)
{
  __shared__ float aC[16 * 40];   // c_attn [k][d], stride 40 -> disjoint bank sets
  __shared__ float aR[16 * 40];   // r_attn [k][d]
  __shared__ float lwr[32 * 9], lwc[32 * 9];
  __shared__ float lbr[32], lbc[32];

  const int b = blockIdx.z, h = blockIdx.y, y = blockIdx.x;
  const int tid = threadIdx.x;
  const int lane = tid & 31, lh = lane >> 4, l16 = lane & 15;
  const int wv = tid >> 5;

  const float* aRg = attn + (((size_t)0 * 4 + b) * 8 + h) * 512;  // rgb branch
  const float* aCg = attn + (((size_t)1 * 4 + b) * 8 + h) * 512;  // chm branch
  for (int t = tid; t < 512; t += 256) {
    const int k = t >> 5, d = t & 31;
    aC[k * 40 + d] = aCg[t];
    aR[k * 40 + d] = aRg[t];
  }
  for (int t = tid; t < 288; t += 256) {
    lwr[t] = rpw[(size_t)(32 * h) * 9 + t];
    lwc[t] = cpw[(size_t)(32 * h) * 9 + t];
  }
  if (tid < 32) { lbr[tid] = rpb[32 * h + tid]; lbc[tid] = cpb[32 * h + tid]; }
  __syncthreads();

  const int x  = wv * 16 + l16;
  const int ng = y * WIDTH + x;

  const float* rq = qkv + (((size_t)0 * 4 + b) * OCH + 64 * h) * N_SP;
  const float* cq = qkv + (((size_t)1 * 4 + b) * OCH + 64 * h) * N_SP;

  v8f A0 = {}, A1 = {}, B0 = {}, B1 = {};
#pragma unroll
  for (int ks = 0; ks < 4; ++ks) {
    const int kb = ks * 4 + 2 * lh;
    v2f qf, cf, a;
    qf.x = rq[(size_t)kb * N_SP + ng];  qf.y = rq[(size_t)(kb + 1) * N_SP + ng];
    cf.x = cq[(size_t)kb * N_SP + ng];  cf.y = cq[(size_t)(kb + 1) * N_SP + ng];
    a.x = aC[kb * 40 + l16];       a.y = aC[(kb + 1) * 40 + l16];       A0 = wmma4(a, qf, A0);
    a.x = aC[kb * 40 + 16 + l16];  a.y = aC[(kb + 1) * 40 + 16 + l16];  A1 = wmma4(a, qf, A1);
    a.x = aR[kb * 40 + l16];       a.y = aR[(kb + 1) * 40 + l16];       B0 = wmma4(a, cf, B0);
    a.x = aR[kb * 40 + 16 + l16];  a.y = aR[(kb + 1) * 40 + 16 + l16];  B1 = wmma4(a, cf, B1);
  }

  const float* rv = qkv + (((size_t)0 * 4 + b) * OCH + 64 * h + 32) * N_SP;
  const float* cv = qkv + (((size_t)1 * 4 + b) * OCH + 64 * h + 32) * N_SP;
  float* op = out + ((size_t)b * CCH + 32 * h) * N_SP + ng;

#pragma unroll
  for (int t = 0; t < 2; ++t) {
#pragma unroll
    for (int r = 0; r < 8; ++r) {
      const int d = t * 16 + r + 8 * lh;
      float val = (t == 0) ? (A0[r] + B0[r]) : (A1[r] + B1[r]);
      val += lbr[d] + lbc[d];
      const float* rvd = rv + (size_t)d * N_SP;
      const float* cvd = cv + (size_t)d * N_SP;
      const float* wr = &lwr[d * 9];
      const float* wc = &lwc[d * 9];
#pragma unroll
      for (int ky = 0; ky < 3; ++ky) {
        const int yy = y + ky - 1;
        if (yy < 0 || yy >= 128) continue;
#pragma unroll
        for (int kx = 0; kx < 3; ++kx) {
          const int xx = x + kx - 1;
          if (xx < 0 || xx >= 128) continue;
          const int sp = yy * WIDTH + xx;
          val += rvd[sp] * wr[ky * 3 + kx];
          val += cvd[sp] * wc[ky * 3 + kx];
        }
      }
      op[(size_t)d * N_SP] = val;
    }
  }
}

// ---------------------------------------------------------------------------
extern "C" void kernel_launch(void* const* d_in, const int* in_sizes, int n_in,
                              void* d_out, int out_size, void* d_ws, size_t ws_size,
                              hipStream_t stream) {
  const float* rgb = (const float*)d_in[0];
  const float* chm = (const float*)d_in[1];
  const float* rw  = (const float*)d_in[2];
  const float* rb  = (const float*)d_in[3];
  const float* cw  = (const float*)d_in[4];
  const float* cb  = (const float*)d_in[5];
  const float* rpw = (const float*)d_in[6];
  const float* rpb = (const float*)d_in[7];
  const float* cpw = (const float*)d_in[8];
  const float* cpb = (const float*)d_in[9];
  float* out = (float*)d_out;

  // workspace layout (floats): qkv[2*4*512*16384] | part[64*128*512] | attn[64*512]
  float* qkv  = (float*)d_ws;
  float* part = qkv + (size_t)2 * 4 * OCH * N_SP;   // 67,108,864 floats
  float* attn = part + (size_t)64 * 128 * 512;      //  4,194,304 floats

  qkv_kvred_kernel<<<dim3(128, 8, 8), 256, 0, stream>>>(rgb, chm, rw, rb, cw, cb,
                                                        qkv, part);
  reduce_softmax_kernel<<<64, 256, 0, stream>>>(part, attn);
  output_kernel<<<dim3(128, 8, 4), 256, 0, stream>>>(qkv, attn, rpw, rpb, cpw, cpb,
                                                     out);
}